// Attention_3831110828381
// MI455X (gfx1250) — compile-verified
//
#include <hip/hip_runtime.h>
#include <math.h>

// Problem constants (match reference)
#define B_    32
#define H_    32
#define KVH_  8
#define G_    4      // GQA group size = H/KVH
#define DH_   128
#define BS_   16
#define NBLK_ 128
#define SCALE_ 0.08838834764831845f  // 128^-0.5

#define KVROW_ (KVH_ * DH_)          // 1024 floats between consecutive slots

typedef __attribute__((ext_vector_type(2))) float v2f;
typedef __attribute__((ext_vector_type(8))) float v8f;

#define QS 130   // q_lds row stride (floats), padded for LDS bank spread
#define STR 17   // score-transpose row stride (floats), padded

__device__ __forceinline__ v8f wmma4(v2f a, v2f b, v8f c) {
  // D(16x16,f32) = A(16x4,f32) * B(4x16,f32) + C
  return __builtin_amdgcn_wmma_f32_16x16x4_f32(
      /*neg_a=*/false, a, /*neg_b=*/false, b,
      /*c_mod=*/(short)0, c, /*reuse_a=*/false, /*reuse_b=*/false);
}

__global__ __launch_bounds__(256) void paged_decode_attn(
    const float* __restrict__ q,
    const float* __restrict__ k_new,
    const float* __restrict__ v_new,
    const float* __restrict__ k_cache,
    const float* __restrict__ v_cache,
    const int*  __restrict__ block_tables,
    const int*  __restrict__ context_lens,
    float* __restrict__ out)
{
  __shared__ float q_lds[16 * QS];          // scaled Q, rows >= G_ zeroed
  __shared__ float sT[8][16 * STR];         // per-wave P transpose buffer
  __shared__ float o_lds[8][G_][DH_];       // per-wave output partials
  __shared__ float m_lds[8][G_];            // per-wave running max
  __shared__ float s_lds[8][G_];            // per-wave running sum

  const int tid  = threadIdx.x;
  const int w    = tid >> 5;                // wave id 0..7 (wave32)
  const int lane = tid & 31;
  const int g16  = lane & 15;               // column/row index within 16
  const int hi   = lane >> 4;               // upper-half-of-wave flag

  const int bkv = blockIdx.x;
  const int b   = bkv / KVH_;
  const int kvh = bkv % KVH_;

  const int n  = context_lens[b];           // valid kv length (incl. new token)
  const int nc = n - 1;                     // cache-resident positions: [0, nc)
  const int* bt = block_tables + b * NBLK_;

  // ---- stage Q*SCALE into LDS (rows 4..15 zero -> clean unused WMMA cols) ----
  for (int i = tid; i < 16 * QS; i += 256) q_lds[i] = 0.0f;
  __syncthreads();
  for (int i = tid; i < G_ * DH_; i += 256) {
    int g = i >> 7, d = i & 127;
    q_lds[g * QS + d] = q[((size_t)b * H_ + (size_t)kvh * G_ + g) * DH_ + d] * SCALE_;
  }
  __syncthreads();

  const float* knew_row = k_new + ((size_t)b * KVH_ + kvh) * DH_;
  const float* vnew_row = v_new + ((size_t)b * KVH_ + kvh) * DH_;

  float m_run = -INFINITY;
  float s_run = 0.0f;
  v8f o_acc[8] = {};                        // O[g, d] tiles: D rows 0..3 valid

  const int ntc = (nc + BS_ - 1) / BS_;     // tiles over cache positions

  for (int t = w; t < ntc; t += 8) {
    const int blk = __builtin_amdgcn_readfirstlane(bt[t]);

    // prefetch next tile's K/V rows (half-wave covers 256B each of 512B row)
    if (t + 8 < ntc) {
      const int blk2 = __builtin_amdgcn_readfirstlane(bt[t + 8]);
      const size_t roff = (((size_t)blk2 * BS_ + g16) * KVH_ + kvh) * DH_;
      const char* kp = (const char*)(k_cache + roff);
      const char* vp = (const char*)(v_cache + roff);
      __builtin_prefetch(kp + hi * 256, 0, 0);
      __builtin_prefetch(kp + hi * 256 + 128, 0, 0);
      __builtin_prefetch(vp + hi * 256, 0, 0);
      __builtin_prefetch(vp + hi * 256 + 128, 0, 0);
    }

    // ---- K tile: A-matrix layout (lane row = key l = g16, float2/chunk) ----
    const float* kb = k_cache + (((size_t)blk * BS_ + g16) * KVH_ + kvh) * DH_;
    v2f ka[32];
    #pragma unroll
    for (int c = 0; c < 32; ++c)
      ka[c] = *(const v2f*)(kb + 4 * c + 2 * hi);

    // ---- S[l, g] = K · Q^T (scale folded into Q) ----
    v8f sacc = {};
    #pragma unroll
    for (int c = 0; c < 32; ++c) {
      v2f qv = *(const v2f*)(&q_lds[g16 * QS + 4 * c + 2 * hi]);
      sacc = wmma4(ka[c], qv, sacc);
    }

    // ---- mask + online softmax; lane owns column g = g16 ----
    const int base = t * BS_;
    float p[8];
    float tmax = -INFINITY;
    #pragma unroll
    for (int j = 0; j < 8; ++j) {
      int l = j + 8 * hi;                   // key row held by this VGPR slot
      float sc = (base + l < nc) ? sacc[j] : -INFINITY;
      p[j] = sc;
      tmax = fmaxf(tmax, sc);
    }
    tmax = fmaxf(tmax, __shfl_xor(tmax, 16, 32));
    const float m_new = fmaxf(m_run, tmax);     // finite: tile has >=1 valid key
    const float corr  = __expf(m_run - m_new);  // 0 on first tile
    float psum = 0.0f;
    #pragma unroll
    for (int j = 0; j < 8; ++j) {
      p[j] = __expf(p[j] - m_new);
      psum += p[j];
    }
    psum += __shfl_xor(psum, 16, 32);
    s_run = s_run * corr + psum;
    m_run = m_new;

    // rescale accumulators: factor for row g comes from column-lane g
    const float f0 = __shfl(corr, 0, 32);
    const float f1 = __shfl(corr, 1, 32);
    const float f2 = __shfl(corr, 2, 32);
    const float f3 = __shfl(corr, 3, 32);
    #pragma unroll
    for (int m = 0; m < 8; ++m) {
      o_acc[m][0] *= f0; o_acc[m][1] *= f1;
      o_acc[m][2] *= f2; o_acc[m][3] *= f3;
    }

    // ---- transpose P (l,g)->(g,l) via wave-private LDS tile ----
    #pragma unroll
    for (int j = 0; j < 8; ++j)
      sT[w][(j + 8 * hi) * STR + g16] = p[j];

    // ---- O[g,d] += P · V : A = P chunk, B = V columns (branch-free addrs) ----
    const float* vblk = v_cache + ((size_t)blk * BS_ * KVH_ + kvh) * DH_;
    #pragma unroll
    for (int c = 0; c < 4; ++c) {
      const int l0 = 4 * c + 2 * hi;
      const float* vp0 = vblk + (size_t)l0 * KVROW_;
      const float* vp1 = vp0 + KVROW_;
      v2f pa;
      pa.x = sT[w][l0 * STR + g16];
      pa.y = sT[w][(l0 + 1) * STR + g16];
      #pragma unroll
      for (int m = 0; m < 8; ++m) {
        v2f vb;
        vb.x = vp0[m * 16 + g16];
        vb.y = vp1[m * 16 + g16];
        o_acc[m] = wmma4(pa, vb, o_acc[m]);
      }
    }
  }

  // ---- wave 0 folds in the new token (position n-1): K/V from k_new/v_new ----
  if (w == 0) {
    float dot[G_];
    #pragma unroll
    for (int g = 0; g < G_; ++g) {
      float acc_d = 0.0f;
      #pragma unroll
      for (int it = 0; it < DH_ / 32; ++it)
        acc_d += q_lds[g * QS + lane + it * 32] * knew_row[lane + it * 32];
      #pragma unroll
      for (int off = 16; off > 0; off >>= 1)
        acc_d += __shfl_xor(acc_d, off, 32);
      dot[g] = acc_d;                       // every lane holds the full dot
    }
    const float sc_new = (g16 == 0) ? dot[0] : (g16 == 1) ? dot[1] :
                         (g16 == 2) ? dot[2] : (g16 == 3) ? dot[3] : -INFINITY;
    const float m2   = fmaxf(m_run, sc_new);    // finite for column lanes g<4
    const float corr = __expf(m_run - m2);
    const float pn   = __expf(sc_new - m2);
    s_run = s_run * corr + pn;
    m_run = m2;
    const float c0 = __shfl(corr, 0, 32), c1 = __shfl(corr, 1, 32),
                c2 = __shfl(corr, 2, 32), c3 = __shfl(corr, 3, 32);
    const float p0 = __shfl(pn, 0, 32),   p1 = __shfl(pn, 1, 32),
                p2 = __shfl(pn, 2, 32),   p3 = __shfl(pn, 3, 32);
    #pragma unroll
    for (int m = 0; m < 8; ++m) {
      const float vn = vnew_row[m * 16 + g16];
      o_acc[m][0] = o_acc[m][0] * c0 + p0 * vn;
      o_acc[m][1] = o_acc[m][1] * c1 + p1 * vn;
      o_acc[m][2] = o_acc[m][2] * c2 + p2 * vn;
      o_acc[m][3] = o_acc[m][3] * c3 + p3 * vn;
    }
  }

  // ---- publish wave partials ----
  if (lane < G_) {
    m_lds[w][lane] = m_run;
    s_lds[w][lane] = s_run;
  }
  if (lane < 16) {
    #pragma unroll
    for (int m = 0; m < 8; ++m)
      #pragma unroll
      for (int j = 0; j < G_; ++j)
        o_lds[w][j][m * 16 + lane] = o_acc[m][j];
  }
  __syncthreads();

  // ---- log-sum-exp combine of the 8 wave partials, normalize, store ----
  for (int i = tid; i < G_ * DH_; i += 256) {
    const int g = i >> 7, d = i & 127;
    float Mg = -INFINITY;
    #pragma unroll
    for (int ww = 0; ww < 8; ++ww) Mg = fmaxf(Mg, m_lds[ww][g]);
    float T = 0.0f, oo = 0.0f;
    #pragma unroll
    for (int ww = 0; ww < 8; ++ww) {
      const float al = __expf(m_lds[ww][g] - Mg);   // 0 for idle waves
      T  += s_lds[ww][g] * al;
      oo += o_lds[ww][g][d] * al;
    }
    out[((size_t)b * H_ + (size_t)kvh * G_ + g) * DH_ + d] = oo / T;
  }
}

extern "C" void kernel_launch(void* const* d_in, const int* in_sizes, int n_in,
                              void* d_out, int out_size, void* d_ws, size_t ws_size,
                              hipStream_t stream) {
  (void)in_sizes; (void)n_in; (void)out_size; (void)d_ws; (void)ws_size;
  const float* q          = (const float*)d_in[0];
  const float* k_new      = (const float*)d_in[1];
  const float* v_new      = (const float*)d_in[2];
  const float* k_cache    = (const float*)d_in[3];
  const float* v_cache    = (const float*)d_in[4];
  const int*   block_tab  = (const int*)d_in[5];
  const int*   ctx_lens   = (const int*)d_in[6];
  // d_in[7] = slot_mapping: unused — the new token is always position n-1,
  // handled directly from k_new/v_new (cache is never mutated).
  float*       out        = (float*)d_out;

  paged_decode_attn<<<dim3(B_ * KVH_), dim3(256), 0, stream>>>(
      q, k_new, v_new, k_cache, v_cache, block_tab, ctx_lens, out);
}